// MSELoss_6820408066430
// MI455X (gfx1250) — compile-verified
//
#include <hip/hip_runtime.h>

// MSE loss over B=16384, D=4096 fp32 tensors.
// Bandwidth-bound: 537 MB of single-use reads -> NT B128 loads, VGPR
// accumulation, deterministic two-pass tree reduction. WMMA f32 16x16x4
// (all-ones A) performs the fp32 wave-level sum.

typedef float f4  __attribute__((ext_vector_type(4)));
typedef float v2f __attribute__((ext_vector_type(2)));
typedef float v8f __attribute__((ext_vector_type(8)));

// Reduce one fp32 value per lane to the wave total (returned in all lanes).
// Uses V_WMMA_F32_16X16X4_F32 with A = ones(16x4): D[m][n] = sum_k B[k][n].
// Lane accs live in B's VGPR0 (VGPR1 = 0); the (vgpr,lane)->(k,n) mapping is
// a bijection, so sum_n colsum(n) == sum of all lane accs, layout-independent.
__device__ __forceinline__ float wave_reduce_wmma(float acc) {
    v2f a;  a.x = 1.0f; a.y = 1.0f;      // A all-ones: layout-agnostic
    v2f b;  b.x = acc;  b.y = 0.0f;      // lane accs in B VGPR0
    v8f c = {};
    v8f d = __builtin_amdgcn_wmma_f32_16x16x4_f32(
        /*neg_a=*/false, a, /*neg_b=*/false, b,
        /*c_mod=*/(short)0, c, /*reuse_a=*/false, /*reuse_b=*/false);
    // d[0] = colsum(lane & 15), duplicated across 16-lane halves.
    float r = d[0];
    r += __shfl_xor(r, 1);
    r += __shfl_xor(r, 2);
    r += __shfl_xor(r, 4);
    r += __shfl_xor(r, 8);
    return r;  // wave total in every lane
}

// Block reduce (256 threads = 8 waves). Returns total to thread 0.
__device__ __forceinline__ float block_reduce(float acc) {
    __shared__ float sm[8];
    float w = wave_reduce_wmma(acc);
    const int wave = threadIdx.x >> 5;
    if ((threadIdx.x & 31) == 0) sm[wave] = w;
    __syncthreads();
    float t = 0.0f;
    if (threadIdx.x == 0) {
        #pragma unroll
        for (int i = 0; i < 8; ++i) t += sm[i];
    }
    return t;
}

__global__ void mse_partial_kernel(const float* __restrict__ yhat,
                                   const float* __restrict__ y,
                                   float* __restrict__ partials,
                                   unsigned int n4 /* n/4 */,
                                   unsigned int ntail,
                                   const float* __restrict__ yhat_tail,
                                   const float* __restrict__ y_tail) {
    const f4* a4 = (const f4*)yhat;
    const f4* b4 = (const f4*)y;
    unsigned int i      = blockIdx.x * blockDim.x + threadIdx.x;
    unsigned int stride = gridDim.x * blockDim.x;

    float acc = 0.0f;
    #pragma unroll 4
    for (; i < n4; i += stride) {
        f4 va = __builtin_nontemporal_load(a4 + i);   // global_load_b128, TH=NT
        f4 vb = __builtin_nontemporal_load(b4 + i);
        f4 df = va - vb;
        acc = fmaf(df.x, df.x, acc);
        acc = fmaf(df.y, df.y, acc);
        acc = fmaf(df.z, df.z, acc);
        acc = fmaf(df.w, df.w, acc);
    }
    // Scalar tail (n % 4 != 0) — handled by thread 0 of block 0 (n is a
    // multiple of 4 for this problem; kept for generality).
    if (blockIdx.x == 0 && threadIdx.x == 0) {
        for (unsigned int t = 0; t < ntail; ++t) {
            float df = yhat_tail[t] - y_tail[t];
            acc = fmaf(df, df, acc);
        }
    }

    float total = block_reduce(acc);
    if (threadIdx.x == 0) partials[blockIdx.x] = total;
}

__global__ void mse_final_kernel(const float* __restrict__ partials,
                                 unsigned int nparts,
                                 float* __restrict__ out,
                                 unsigned long long n) {
    float acc = 0.0f;
    for (unsigned int i = threadIdx.x; i < nparts; i += blockDim.x)
        acc += partials[i];
    float total = block_reduce(acc);
    if (threadIdx.x == 0)
        out[0] = (float)((double)total / (double)n);
}

extern "C" void kernel_launch(void* const* d_in, const int* in_sizes, int n_in,
                              void* d_out, int out_size, void* d_ws, size_t ws_size,
                              hipStream_t stream) {
    const float* yhat = (const float*)d_in[0];
    const float* y    = (const float*)d_in[1];
    float* out        = (float*)d_out;
    float* partials   = (float*)d_ws;

    const unsigned long long n = (unsigned long long)in_sizes[0];
    const unsigned int n4    = (unsigned int)(n >> 2);
    const unsigned int ntail = (unsigned int)(n & 3ull);

    const int THREADS = 256;               // 8 wave32s / block
    int blocks = 2048;                     // 524,288 threads in flight
    if (ws_size < (size_t)blocks * sizeof(float)) {
        blocks = (int)(ws_size / sizeof(float));
        if (blocks < 1) blocks = 1;
    }

    mse_partial_kernel<<<blocks, THREADS, 0, stream>>>(
        yhat, y, partials, n4, ntail, yhat + (n - ntail), y + (n - ntail));
    mse_final_kernel<<<1, THREADS, 0, stream>>>(
        partials, (unsigned int)blocks, out, n);
}